// DPLRSSMBlock_9680856285214
// MI455X (gfx1250) — compile-verified
//
#include <hip/hip_runtime.h>

typedef __attribute__((ext_vector_type(2))) float v2f;
typedef __attribute__((ext_vector_type(4))) float v4f;
typedef __attribute__((ext_vector_type(8))) float v8f;

#define HDIM 8192
#define RDIM 4
#define BDIM 64

// ---------------------------------------------------------------------------
// Kernel 1: t = h @ q   (64 x 8192) @ (8192 x 4) -> (64 x 4)
// One block per batch row; 256 threads strided over K, LDS tree reduction.
// ---------------------------------------------------------------------------
__global__ __launch_bounds__(256)
void dplr_rank_proj_kernel(const float* __restrict__ h,
                           const float* __restrict__ q_vec,
                           float* __restrict__ t) {
    __shared__ float red[256 * RDIM];
    const int b   = blockIdx.x;
    const int tid = threadIdx.x;

    float a0 = 0.f, a1 = 0.f, a2 = 0.f, a3 = 0.f;
    const float* hrow = h + b * HDIM;
    for (int k = tid; k < HDIM; k += 256) {
        const float hv = hrow[k];
        const v4f  qv = *(const v4f*)(q_vec + k * RDIM);  // q_vec is (H, R) row-major
        a0 += hv * qv.x;
        a1 += hv * qv.y;
        a2 += hv * qv.z;
        a3 += hv * qv.w;
    }
    red[tid * RDIM + 0] = a0;
    red[tid * RDIM + 1] = a1;
    red[tid * RDIM + 2] = a2;
    red[tid * RDIM + 3] = a3;
    __syncthreads();

    for (int s = 128; s > 0; s >>= 1) {
        if (tid < s) {
            red[tid * RDIM + 0] += red[(tid + s) * RDIM + 0];
            red[tid * RDIM + 1] += red[(tid + s) * RDIM + 1];
            red[tid * RDIM + 2] += red[(tid + s) * RDIM + 2];
            red[tid * RDIM + 3] += red[(tid + s) * RDIM + 3];
        }
        __syncthreads();
    }
    if (tid < RDIM) t[b * RDIM + tid] = red[tid];
}

// ---------------------------------------------------------------------------
// Kernel 2: out = x @ b_mat + h*a_diag + t @ p^T   via V_WMMA_F32_16X16X4_F32
// Each wave32 owns one 16x16 output tile; K marches in steps of 4.
// FOUR independent accumulator chains (K-steps round-robin) push the WMMA
// accumulator-RAW dependency distance to 4 instructions so the XDL pipe can
// stream back-to-back without s_delay_alu stalls.
// 256 threads = 8 waves: waves 0-3 cover M tiles {0,16,32,48} of n-tile 0,
// waves 4-7 the same M tiles of n-tile 1; block covers a 64x32 output slab.
// ---------------------------------------------------------------------------
__global__ __launch_bounds__(256)
void dplr_gemm_wmma_kernel(const float* __restrict__ x,
                           const float* __restrict__ b_mat,
                           const float* __restrict__ h,
                           const float* __restrict__ a_diag,
                           const float* __restrict__ p_vec,
                           const float* __restrict__ t,
                           float* __restrict__ out) {
    const int lane = threadIdx.x & 31;
    const int wave = threadIdx.x >> 5;            // 0..7
    const int m0   = (wave & 3) * 16;             // M tile base (batch rows)
    const int n0   = blockIdx.x * 32 + (wave >> 2) * 16;  // N tile base (columns)

    const int half = lane >> 4;                   // 0: K=k..k+1, 1: K=k+2..k+3
    const int l    = lane & 15;

    // A fragment source: x[m0+l][k + 2*half .. +1]  (one b64 per lane per step)
    const float* __restrict__ xptr = x + (m0 + l) * HDIM + half * 2;

    // B fragment source: VGPR0 = b_mat[k+2*half][n0+l], VGPR1 = b_mat[k+1+2*half][n0+l]
    const float* __restrict__ brow0 = b_mat + (size_t)(half * 2 + 0) * HDIM + n0 + l;
    const float* __restrict__ brow1 = b_mat + (size_t)(half * 2 + 1) * HDIM + n0 + l;

    v8f acc[4];
#pragma unroll
    for (int c = 0; c < 4; ++c)
        acc[c] = (v8f){0.f, 0.f, 0.f, 0.f, 0.f, 0.f, 0.f, 0.f};

    for (int k0 = 0; k0 < HDIM; k0 += 64) {
        // Speculative prefetch of the b_mat stream a few chunks ahead
        // (silently dropped past the end of the allocation).
        __builtin_prefetch(brow0 + (size_t)(k0 + 256) * HDIM, 0, 1);
        __builtin_prefetch(brow1 + (size_t)(k0 + 256) * HDIM, 0, 1);
#pragma unroll
        for (int k = k0; k < k0 + 64; k += 16) {
#pragma unroll
            for (int c = 0; c < 4; ++c) {      // chain c gets K-step k + 4*c
                const int kc = k + 4 * c;
                v2f afrag = *(const v2f*)(xptr + kc);
                v2f bfrag;
                bfrag.x = brow0[(size_t)kc * HDIM];
                bfrag.y = brow1[(size_t)kc * HDIM];
                acc[c] = __builtin_amdgcn_wmma_f32_16x16x4_f32(
                    false, afrag, false, bfrag, (short)0, acc[c], false, false);
            }
        }
    }

    const v8f accs = (acc[0] + acc[1]) + (acc[2] + acc[3]);

    // Epilogue fused into the C/D VGPR layout:
    //   VGPR v, lanes 0-15  -> (M = m0+v,   N = n0+l)
    //   VGPR v, lanes 16-31 -> (M = m0+v+8, N = n0+l)
    const int   n  = n0 + l;
    const float an = a_diag[n];
    const v4f   pn = *(const v4f*)(p_vec + n * RDIM);  // p_vec is (H, R) row-major

#pragma unroll
    for (int v = 0; v < 8; ++v) {
        const int mm = m0 + v + half * 8;
        const v4f tm = *(const v4f*)(t + mm * RDIM);
        const float corr = h[mm * HDIM + n] * an
                         + tm.x * pn.x + tm.y * pn.y + tm.z * pn.z + tm.w * pn.w;
        out[mm * HDIM + n] = accs[v] + corr;
    }
}

// ---------------------------------------------------------------------------
// Launch
// ---------------------------------------------------------------------------
extern "C" void kernel_launch(void* const* d_in, const int* in_sizes, int n_in,
                              void* d_out, int out_size, void* d_ws, size_t ws_size,
                              hipStream_t stream) {
    (void)in_sizes; (void)n_in; (void)out_size; (void)ws_size;
    const float* h      = (const float*)d_in[0];
    const float* x      = (const float*)d_in[1];
    const float* a_diag = (const float*)d_in[2];
    const float* p_vec  = (const float*)d_in[3];
    const float* q_vec  = (const float*)d_in[4];
    const float* b_mat  = (const float*)d_in[5];
    float* out = (float*)d_out;
    float* t   = (float*)d_ws;   // 64 x 4 floats of scratch

    dplr_rank_proj_kernel<<<BDIM, 256, 0, stream>>>(h, q_vec, t);
    dplr_gemm_wmma_kernel<<<HDIM / 32, 256, 0, stream>>>(x, b_mat, h, a_diag,
                                                         p_vec, t, out);
}